// GMMConv_stash_67508295958898
// MI455X (gfx1250) — compile-verified
//
#include <hip/hip_runtime.h>

typedef float v2f __attribute__((ext_vector_type(2)));
typedef float v8f __attribute__((ext_vector_type(8)));

#define IN_FEATS  64
#define OUT_FEATS 32
#define N_KERNELS 8
#define NKF       (N_KERNELS * OUT_FEATS)   // 256
#define DIM       2

// ---------------------------------------------------------------------------
// Kernel 1: node_feat[N,256] = feat[N,64] @ W^T[64,256], fp32 WMMA 16x16x4.
// One wave per 16(M)x16(N) tile; K=64 -> 16 chained v_wmma_f32_16x16x4_f32.
// A layout (ISA 7.12.2, 32-bit A 16x4): lane<16 holds K={0,1}, lane>=16 K={2,3}.
// B[k][n] = W[n][k] -> same per-lane float2 fetch from W row n.
// ---------------------------------------------------------------------------
__global__ __launch_bounds__(256)
void fc_wmma_kernel(const float* __restrict__ feat,
                    const float* __restrict__ W,
                    float* __restrict__ nodefeat,
                    int n, int mtiles) {
  const int wave = blockIdx.x * (blockDim.x >> 5) + (threadIdx.x >> 5);
  const int lane = threadIdx.x & 31;
  const int tn = wave & 15;        // 16 tiles over 256 output channels
  const int tm = wave >> 4;        // tiles over nodes
  if (tm >= mtiles) return;        // wave-uniform: EXEC stays all-1s below

  const int half = lane >> 4;      // 0: K pair {0,1}; 1: K pair {2,3}
  const int r    = lane & 15;

  int mrow = tm * 16 + r;
  if (mrow >= n) mrow = n - 1;     // clamp reads (n % 16 != 0 case)
  const float* __restrict__ arow = feat + (size_t)mrow * IN_FEATS;
  const float* __restrict__ brow = W + (size_t)(tn * 16 + r) * IN_FEATS;

  v8f c = {};
#pragma unroll
  for (int kk = 0; kk < IN_FEATS / 4; ++kk) {
    const int kb = kk * 4 + half * 2;
    const float2 af = *(const float2*)(arow + kb);
    const float2 bf = *(const float2*)(brow + kb);
    v2f a; a.x = af.x; a.y = af.y;
    v2f b; b.x = bf.x; b.y = bf.y;
    // 8 args: (neg_a, A, neg_b, B, c_mod, C, reuse_a, reuse_b)
    c = __builtin_amdgcn_wmma_f32_16x16x4_f32(
        false, a, false, b, (short)0, c, false, false);
  }

  // D layout: VGPR v -> M = v + half*8 (lanes 16-31 hold M=8..15), N = r
#pragma unroll
  for (int v = 0; v < 8; ++v) {
    const int row = tm * 16 + v + half * 8;
    if (row < n)
      nodefeat[(size_t)row * NKF + tn * 16 + r] = c[v];
  }
}

// ---------------------------------------------------------------------------
// Kernel 2: per-edge Gaussian weights + gather-weighted sum + segment reduce.
// One wave per destination node; lane == output feature (F=32 == wave32).
// node_feat (51.2 MB) lives in the 192 MB L2, so the per-k gathers
// (128 B coalesced per k) are L2-resident.
// ---------------------------------------------------------------------------
__global__ __launch_bounds__(256)
void gmm_edge_kernel(const int* __restrict__ rowptr,
                     const int* __restrict__ colind,
                     const int* __restrict__ permute,
                     const float* __restrict__ pseudo,
                     const float* __restrict__ nodefeat,
                     const float* __restrict__ mu,
                     const float* __restrict__ inv_sigma,
                     const float* __restrict__ bias,
                     float* __restrict__ out,
                     int n) {
  __shared__ float s_mu[N_KERNELS * DIM];
  __shared__ float s_is[N_KERNELS * DIM];
  __shared__ float s_bias[OUT_FEATS];

  const int t = threadIdx.x;
  if (t < N_KERNELS * DIM) { s_mu[t] = mu[t]; s_is[t] = inv_sigma[t]; }
  if (t < OUT_FEATS)       { s_bias[t] = bias[t]; }
  __syncthreads();

  const int node = blockIdx.x * (blockDim.x >> 5) + (t >> 5);
  const int f    = t & 31;
  if (node >= n) return;

  const int start = rowptr[node];
  const int end   = rowptr[node + 1];

  float acc = 0.0f;
  for (int e = start; e < end; ++e) {
    const int src = colind[e];
    const int pe  = permute[e];
    const float p0 = pseudo[(size_t)pe * 2 + 0];
    const float p1 = pseudo[(size_t)pe * 2 + 1];
    const float* __restrict__ nf = nodefeat + (size_t)src * NKF + f;
#pragma unroll
    for (int k = 0; k < N_KERNELS; ++k) {
      const float d0 = (p0 - s_mu[k * 2 + 0]) * s_is[k * 2 + 0];
      const float d1 = (p1 - s_mu[k * 2 + 1]) * s_is[k * 2 + 1];
      const float wk = __expf(-0.5f * (d0 * d0 + d1 * d1));
      acc = fmaf(wk, nf[k * OUT_FEATS], acc);
    }
  }
  out[(size_t)node * OUT_FEATS + f] = acc + s_bias[f];
}

// ---------------------------------------------------------------------------
// Inputs (setup_inputs order):
//  0 rowptr  1 colind  2 colptr  3 rowind  4 permute
//  5 feat    6 pseudo  7 W       8 mu      9 inv_sigma  10 bias
// d_ws: node_feat [N,256] fp32 (51.2 MB for N=50000)
// ---------------------------------------------------------------------------
extern "C" void kernel_launch(void* const* d_in, const int* in_sizes, int n_in,
                              void* d_out, int out_size, void* d_ws, size_t ws_size,
                              hipStream_t stream) {
  const int*   rowptr    = (const int*)d_in[0];
  const int*   colind    = (const int*)d_in[1];
  const int*   permute   = (const int*)d_in[4];
  const float* feat      = (const float*)d_in[5];
  const float* pseudo    = (const float*)d_in[6];
  const float* W         = (const float*)d_in[7];
  const float* mu        = (const float*)d_in[8];
  const float* inv_sigma = (const float*)d_in[9];
  const float* bias      = (const float*)d_in[10];
  float*       out       = (float*)d_out;
  float*       nodefeat  = (float*)d_ws;

  const int n      = in_sizes[0] - 1;       // 50000
  const int mtiles = (n + 15) / 16;         // 3125
  const int waves  = mtiles * 16;           // 16 N-tiles each

  dim3 blk(256);                            // 8 waves per block
  dim3 grid1((waves + 7) / 8);
  hipLaunchKernelGGL(fc_wmma_kernel, grid1, blk, 0, stream,
                     feat, W, nodefeat, n, mtiles);

  dim3 grid2((n + 7) / 8);
  hipLaunchKernelGGL(gmm_edge_kernel, grid2, blk, 0, stream,
                     rowptr, colind, permute, pseudo, nodefeat,
                     mu, inv_sigma, bias, out, n);
}